// SynthesisMLP_46866683134472
// MI455X (gfx1250) — compile-verified
//
#include <hip/hip_runtime.h>
#include <hip/hip_bf16.h>

typedef __attribute__((ext_vector_type(2))) float v2f;
typedef __attribute__((ext_vector_type(8))) float v8f;

#define IN_DIM  4096
#define HID_DIM 8192
#define OUT_DIM 64

// ---------------------------------------------------------------------------
// Layer 1: h[net][c] = relu(b1[c] + sum_k x[k] * W1[k*HID + c])
// grid = (HID/16, 2), block = 128 (4 waves of 32)
// Each block owns one 16-column tile of one network's W1. Each wave owns a
// 1024-wide K slice and reduces it with V_WMMA_F32_16X16X4_F32 (exact fp32):
//   A[m][kk] = W1[k+kk][col+m]   (A 16x4 layout: V0={K0|K2}, V1={K1|K3})
//   B[kk][n] = x[k+kk]           (broadcast across N -> layout-permutation safe)
// D's columns all replicate h_partial[col+m]; row m lives in c[v] with
// m = half*8 + v (lanes 0-15: M=v, lanes 16-31: M=8+v).
// ---------------------------------------------------------------------------
__global__ __launch_bounds__(128)
void mlp46_layer1_wmma(const float* __restrict__ x,
                       const float* __restrict__ W1a, const float* __restrict__ b1a,
                       const float* __restrict__ W1b, const float* __restrict__ b1b,
                       float* __restrict__ h)
{
    const int net = blockIdx.y;
    const float* __restrict__ W1 = net ? W1b : W1a;
    const float* __restrict__ b1 = net ? b1b : b1a;
    float* __restrict__ hout = h + net * HID_DIM;

    __shared__ float xs[IN_DIM];
    __shared__ float red[4][16];

    const int tid = threadIdx.x;            // 0..127
    // Stage x into LDS with wide coalesced loads (x stays hot in L2 anyway).
    {
        const float4* __restrict__ x4 = (const float4*)x;
        float4* xs4 = (float4*)xs;
        for (int i = tid; i < IN_DIM / 4; i += 128)
            xs4[i] = x4[i];
    }
    __syncthreads();

    const int wave = tid >> 5;              // 0..3 (wave32)
    const int lane = tid & 31;
    const int m    = lane & 15;             // M index (output column within tile)
    const int half = lane >> 4;             // 0: K=0,1   1: K=2,3
    const int col  = blockIdx.x * 16;

    // Pointers for this wave's K slice [wave*1024, wave*1024+1024)
    const float* __restrict__ wp = W1 + (size_t)(wave * 1024 + half * 2) * HID_DIM + col + m;
    const float* __restrict__ xp = xs + wave * 1024 + half * 2;

    v8f acc0 = {};
    v8f acc1 = {};

    // 128 iterations, 8 K-values each (2 WMMAs on independent accumulators).
    for (int it = 0; it < 128; ++it) {
        v2f a0 = { wp[0],            wp[HID_DIM] };            // K+0/2, K+1/3
        v2f b0 = { xp[0],            xp[1] };
        v2f a1 = { wp[4 * HID_DIM],  wp[5 * HID_DIM] };        // K+4/6, K+5/7
        v2f b1v = { xp[4],           xp[5] };
        acc0 = __builtin_amdgcn_wmma_f32_16x16x4_f32(
                   false, a0, false, b0, (short)0, acc0, false, false);
        acc1 = __builtin_amdgcn_wmma_f32_16x16x4_f32(
                   false, a1, false, b1v, (short)0, acc1, false, false);
        wp += 8 * HID_DIM;
        xp += 8;
    }

#pragma unroll
    for (int v = 0; v < 8; ++v)
        acc0[v] += acc1[v];

    // All N columns of D are identical; lane with N==0 in each half publishes.
    if (m == 0) {
#pragma unroll
        for (int v = 0; v < 8; ++v)
            red[wave][half * 8 + v] = acc0[v];
    }
    __syncthreads();

    if (tid < 16) {
        float s = red[0][tid] + red[1][tid] + red[2][tid] + red[3][tid];
        s += b1[col + tid];
        hout[col + tid] = s > 0.0f ? s : 0.0f;
    }
}

// ---------------------------------------------------------------------------
// Layer 2 partial sums: partial[(net*64 + chunk)*64 + j] =
//     sum_{k in chunk} h[net][k] * W2[k*64 + j]
// grid = (64, 2), block = 256: j = tid&63 (coalesced 64-wide row reads),
// sub = tid>>6 splits the 128-row chunk 4 ways, reduced via LDS.
// ---------------------------------------------------------------------------
__global__ __launch_bounds__(256)
void mlp46_layer2_partial(const float* __restrict__ h,
                          const float* __restrict__ W2a,
                          const float* __restrict__ W2b,
                          float* __restrict__ partial)
{
    const int net   = blockIdx.y;
    const int chunk = blockIdx.x;                 // 0..63 -> 128 rows each
    const float* __restrict__ W2 = net ? W2b : W2a;
    const float* __restrict__ hp = h + net * HID_DIM;

    const int tid = threadIdx.x;                  // 0..255
    const int j   = tid & 63;
    const int sub = tid >> 6;                     // 0..3
    const int k0  = chunk * 128 + sub * 32;

    float acc = 0.0f;
#pragma unroll
    for (int i = 0; i < 32; ++i) {
        const int k = k0 + i;
        acc += hp[k] * W2[(size_t)k * OUT_DIM + j];
    }

    __shared__ float red[4][64];
    red[sub][j] = acc;
    __syncthreads();
    if (sub == 0) {
        float s = red[0][j] + red[1][j] + red[2][j] + red[3][j];
        partial[(net * 64 + chunk) * 64 + j] = s;
    }
}

// ---------------------------------------------------------------------------
// Finalize: reduce 64 chunk partials per (net, j), add bias, ReLU, and apply
// the exact >0 / ==0 combine rule. Single block => deterministic.
// ---------------------------------------------------------------------------
__global__ __launch_bounds__(128)
void mlp46_finalize(const float* __restrict__ partial,
                    const float* __restrict__ b2a,
                    const float* __restrict__ b2b,
                    float* __restrict__ out)
{
    const int tid = threadIdx.x;                  // 0..127
    const int net = tid >> 6;
    const int j   = tid & 63;
    const float* __restrict__ b2 = net ? b2b : b2a;

    float s = b2[j];
    for (int c = 0; c < 64; ++c)
        s += partial[(net * 64 + c) * 64 + j];
    s = s > 0.0f ? s : 0.0f;                      // ReLU -> exactly >0 or ==0

    __shared__ float o[2][64];
    o[net][j] = s;
    __syncthreads();

    if (tid < 64) {
        const float o1 = o[0][tid];
        const float o2 = o[1][tid];
        float r;
        if (o1 > 0.0f && o2 > 0.0f)      r = 0.5f * (o1 + o2);
        else if (o1 > 0.0f)              r = o1;   // o2 == 0
        else if (o2 > 0.0f)              r = o2;   // o1 == 0
        else                             r = 0.0f;
        out[tid] = r;
    }
}

extern "C" void kernel_launch(void* const* d_in, const int* in_sizes, int n_in,
                              void* d_out, int out_size, void* d_ws, size_t ws_size,
                              hipStream_t stream) {
    (void)in_sizes; (void)n_in; (void)out_size; (void)ws_size;
    const float* x   = (const float*)d_in[0];
    const float* W1a = (const float*)d_in[1];
    const float* b1a = (const float*)d_in[2];
    const float* W2a = (const float*)d_in[3];
    const float* b2a = (const float*)d_in[4];
    const float* W1b = (const float*)d_in[5];
    const float* b1b = (const float*)d_in[6];
    const float* W2b = (const float*)d_in[7];
    const float* b2b = (const float*)d_in[8];

    float* h       = (float*)d_ws;                 // 2*8192 floats = 64 KB
    float* partial = h + 2 * HID_DIM;              // 128*64 floats = 32 KB
    float* out     = (float*)d_out;                // 64 floats

    mlp46_layer1_wmma<<<dim3(HID_DIM / 16, 2), 128, 0, stream>>>(
        x, W1a, b1a, W1b, b1b, h);
    mlp46_layer2_partial<<<dim3(64, 2), 256, 0, stream>>>(
        h, W2a, W2b, partial);
    mlp46_finalize<<<1, 128, 0, stream>>>(partial, b2a, b2b, out);
}